// GraphAttentionLayer_55353538511014
// MI455X (gfx1250) — compile-verified
//
#include <hip/hip_runtime.h>
#include <math.h>

typedef float v2f __attribute__((ext_vector_type(2)));
typedef float v8f __attribute__((ext_vector_type(8)));

#define HEADS 8
#define CH 32
#define F 256            // HEADS*CH == IN
#define NEG_SLOPE 0.2f
#define LN_EPS 1e-5f

#define LDS_STRIDE 260   // 256 + 4 pad: keeps 16B alignment (1040B rows) and
                         // spreads the 16 row-strided lanes over distinct banks

// ---- order-preserving float <-> unsigned key (for atomicMax on floats) ----
__device__ __forceinline__ unsigned f2key(float f) {
  unsigned b = __float_as_uint(f);
  return (b & 0x80000000u) ? ~b : (b | 0x80000000u);
}
__device__ __forceinline__ float key2f(unsigned k) {
  unsigned b = (k & 0x80000000u) ? (k & 0x7fffffffu) : ~k;
  return __uint_as_float(b);
}

// ---------------- init: zero out accumulator + denom, seed segmax ----------
__global__ void gat_init_kernel(float* __restrict__ out, float* __restrict__ denom,
                                unsigned* __restrict__ segmax, int n_out, int n_nh) {
  int stride = gridDim.x * blockDim.x;
  for (int t = blockIdx.x * blockDim.x + threadIdx.x; t < n_out; t += stride) out[t] = 0.f;
  for (int t = blockIdx.x * blockDim.x + threadIdx.x; t < n_nh; t += stride) {
    denom[t] = 0.f;
    segmax[t] = 0x007FFFFFu;   // f2key(-inf)
  }
}

// ---------------- projection: xl = x @ W  via V_WMMA_F32_16X16X4_F32 -------
// 512 threads = 16 waves per block; block covers a 16-row x 256-col tile.
// The 16x256 A tile is staged once into LDS (shared by all 16 waves); each
// wave computes one 16x16 output tile with 64 K=4 WMMAs, A from LDS (ds_load),
// B from global (L2-resident W).
__global__ void __launch_bounds__(512)
gat_gemm_wmma_kernel(const float* __restrict__ x, const float* __restrict__ W,
                     float* __restrict__ xl, int N) {
  __shared__ float lds_a[16 * LDS_STRIDE];

  const int tid  = threadIdx.x;
  const int lane = tid & 31;
  const int wave = tid >> 5;                // 0..15 -> column tile
  const int row0 = blockIdx.x * 16;
  const int col0 = wave * 16;

  // ---- cooperative A-tile stage: thread t loads 8 floats (two float4) ----
  {
    int r = tid >> 5;                       // 0..15
    int c = (tid & 31) * 8;                 // 0..248
    int grow = row0 + r;
    if (grow >= N) grow = N - 1;            // clamp (keeps EXEC full later)
    const float4* __restrict__ src = (const float4*)(x + (size_t)grow * F + c);
    float4* dst = (float4*)(&lds_a[r * LDS_STRIDE + c]);
    dst[0] = src[0];
    dst[1] = src[1];
  }
  __syncthreads();

  // A fragment mapping (16x4 f32): lane m = lane%16, K pair = (lane/16)*2
  const int m    = lane & 15;
  const int koff = (lane >> 4) * 2;         // 0 or 2
  const float* __restrict__ arow = &lds_a[m * LDS_STRIDE];

  // B fragment mapping (4x16 f32): lane n = lane%16, K pair = (lane/16)*2
  const int n = lane & 15;

  v8f acc = {0.f, 0.f, 0.f, 0.f, 0.f, 0.f, 0.f, 0.f};
  for (int k = 0; k < F; k += 4) {
    v2f a = *(const v2f*)(arow + k + koff);         // ds_load_b64
    v2f b;
    b.x = W[(size_t)(k + koff) * F + col0 + n];
    b.y = W[(size_t)(k + koff + 1) * F + col0 + n];
    acc = __builtin_amdgcn_wmma_f32_16x16x4_f32(
        /*neg_a=*/false, a, /*neg_b=*/false, b,
        /*c_mod=*/(short)0, acc, /*reuse_a=*/false, /*reuse_b=*/false);
  }

  // C/D layout: VGPR j -> row (j + (lane/16)*8), col = lane%16
  const int rbase = row0 + ((lane >> 4) * 8);
  float* __restrict__ dst = xl + (size_t)rbase * F + col0 + n;
  if (row0 + 16 <= N) {
    // block-uniform fast path: straight stores, no exec-mask dance
#pragma unroll
    for (int j = 0; j < 8; ++j) dst[(size_t)j * F] = acc[j];
  } else {
#pragma unroll
    for (int j = 0; j < 8; ++j) {
      if (rbase + j < N) dst[(size_t)j * F] = acc[j];
    }
  }
}

// ---------------- per-node attention half-logits ---------------------------
__global__ void gat_att_kernel(const float* __restrict__ xl,
                               const float* __restrict__ att_src,
                               const float* __restrict__ att_dst,
                               float* __restrict__ a_src, float* __restrict__ a_dst,
                               int NH) {
  int t = blockIdx.x * blockDim.x + threadIdx.x;
  if (t >= NH) return;
  int nidx = t >> 3, h = t & 7;
  const float* __restrict__ xr = xl + (size_t)nidx * F + h * CH;
  const float* __restrict__ as = att_src + h * CH;
  const float* __restrict__ ad = att_dst + h * CH;
  float s = 0.f, d = 0.f;
#pragma unroll
  for (int c = 0; c < CH; ++c) {
    float v = xr[c];
    s += v * as[c];
    d += v * ad[c];
  }
  a_src[t] = s;
  a_dst[t] = d;
}

__device__ __forceinline__ void edge_nodes(const int* __restrict__ ei, int e, int E,
                                           int& s, int& d) {
  if (e < E) { s = ei[e]; d = ei[E + e]; }
  else       { s = d = e - E; }             // self loops
}

__device__ __forceinline__ float leaky(float v) {
  return (v >= 0.f) ? v : v * NEG_SLOPE;
}

// ---------------- pass A: segment max of edge logits -----------------------
__global__ void gat_edge_max_kernel(const int* __restrict__ ei,
                                    const float* __restrict__ a_src,
                                    const float* __restrict__ a_dst,
                                    unsigned* __restrict__ segmax, int E, int N) {
  int t = blockIdx.x * blockDim.x + threadIdx.x;
  int total = (E + N) * HEADS;
  if (t >= total) return;
  int e = t >> 3, h = t & 7;
  int s, d;
  edge_nodes(ei, e, E, s, d);
  float logit = leaky(a_src[s * HEADS + h] + a_dst[d * HEADS + h]);
  atomicMax(&segmax[d * HEADS + h], f2key(logit));
}

// ---------------- pass B: segment sum of exp(logit - max) ------------------
__global__ void gat_edge_sum_kernel(const int* __restrict__ ei,
                                    const float* __restrict__ a_src,
                                    const float* __restrict__ a_dst,
                                    const unsigned* __restrict__ segmax,
                                    float* __restrict__ denom, int E, int N) {
  int t = blockIdx.x * blockDim.x + threadIdx.x;
  int total = (E + N) * HEADS;
  if (t >= total) return;
  int e = t >> 3, h = t & 7;
  int s, d;
  edge_nodes(ei, e, E, s, d);
  float logit = leaky(a_src[s * HEADS + h] + a_dst[d * HEADS + h]);
  float mx = key2f(segmax[d * HEADS + h]);
  atomicAdd(&denom[d * HEADS + h], expf(logit - mx));
}

// ---------------- pass C: weighted scatter-aggregate -----------------------
// One 256-thread block per edge; thread = feature index (h = tid/32).
// xl (51MB) and out (51MB) both fit the 192MB L2 -> L2-resident traffic.
__global__ void __launch_bounds__(256)
gat_edge_agg_kernel(const int* __restrict__ ei, const float* __restrict__ xl,
                    const float* __restrict__ a_src, const float* __restrict__ a_dst,
                    const unsigned* __restrict__ segmax, const float* __restrict__ denom,
                    float* __restrict__ out, int E, int N) {
  int e = blockIdx.x;
  int tid = threadIdx.x;
  int h = tid >> 5;
  int s, d;
  edge_nodes(ei, e, E, s, d);
  float logit = leaky(a_src[s * HEADS + h] + a_dst[d * HEADS + h]);
  float mx = key2f(segmax[d * HEADS + h]);
  float alpha = expf(logit - mx) / denom[d * HEADS + h];
  float msg = xl[(size_t)s * F + tid] * alpha;
  atomicAdd(&out[(size_t)d * F + tid], msg);
}

// ---------------- LayerNorm (in place over d_out rows) ---------------------
__global__ void __launch_bounds__(256)
gat_ln_kernel(float* __restrict__ out, const float* __restrict__ bias,
              const float* __restrict__ gamma, const float* __restrict__ beta, int N) {
  __shared__ float red[256];
  int row = blockIdx.x;
  int tid = threadIdx.x;
  float v = out[(size_t)row * F + tid] + bias[tid];

  red[tid] = v;
  __syncthreads();
  for (int s = 128; s > 0; s >>= 1) {
    if (tid < s) red[tid] += red[tid + s];
    __syncthreads();
  }
  float mean = red[0] * (1.0f / F);
  __syncthreads();

  float dv = v - mean;
  red[tid] = dv * dv;
  __syncthreads();
  for (int s = 128; s > 0; s >>= 1) {
    if (tid < s) red[tid] += red[tid + s];
    __syncthreads();
  }
  float var = red[0] * (1.0f / F);

  out[(size_t)row * F + tid] = dv * rsqrtf(var + LN_EPS) * gamma[tid] + beta[tid];
}

// ---------------------------------------------------------------------------
extern "C" void kernel_launch(void* const* d_in, const int* in_sizes, int n_in,
                              void* d_out, int out_size, void* d_ws, size_t ws_size,
                              hipStream_t stream) {
  const float* x       = (const float*)d_in[0];
  const int*   ei      = (const int*)  d_in[1];
  const float* W       = (const float*)d_in[2];
  const float* att_src = (const float*)d_in[3];
  const float* att_dst = (const float*)d_in[4];
  const float* bias    = (const float*)d_in[5];
  const float* gamma   = (const float*)d_in[6];
  const float* beta    = (const float*)d_in[7];
  float* out = (float*)d_out;

  const int N = in_sizes[0] / F;
  const int E = in_sizes[1] / 2;

  char* ws = (char*)d_ws;
  float*    xl     = (float*)ws;    ws += (size_t)N * F * sizeof(float);
  float*    a_src  = (float*)ws;    ws += (size_t)N * HEADS * sizeof(float);
  float*    a_dst  = (float*)ws;    ws += (size_t)N * HEADS * sizeof(float);
  unsigned* segmax = (unsigned*)ws; ws += (size_t)N * HEADS * sizeof(unsigned);
  float*    denom  = (float*)ws;

  // 0) init accumulators (d_out is poisoned by the harness -> must zero each call)
  gat_init_kernel<<<2048, 256, 0, stream>>>(out, denom, segmax, N * F, N * HEADS);

  // 1) projection via fp32 WMMA (LDS-staged A tile)
  gat_gemm_wmma_kernel<<<(N + 15) / 16, 512, 0, stream>>>(x, W, xl, N);

  // 2) per-node half-logits
  {
    int NH = N * HEADS;
    gat_att_kernel<<<(NH + 255) / 256, 256, 0, stream>>>(xl, att_src, att_dst,
                                                         a_src, a_dst, NH);
  }

  // 3) segment softmax (3 passes) + aggregation
  {
    int total = (E + N) * HEADS;
    int blocks = (total + 255) / 256;
    gat_edge_max_kernel<<<blocks, 256, 0, stream>>>(ei, a_src, a_dst, segmax, E, N);
    gat_edge_sum_kernel<<<blocks, 256, 0, stream>>>(ei, a_src, a_dst, segmax, denom, E, N);
    gat_edge_agg_kernel<<<E + N, 256, 0, stream>>>(ei, xl, a_src, a_dst, segmax, denom,
                                                   out, E, N);
  }

  // 4) bias + LayerNorm in place
  gat_ln_kernel<<<N, 256, 0, stream>>>(out, bias, gamma, beta, N);
}